// MentionRepresentation_73598559584945
// MI455X (gfx1250) — compile-verified
//
#include <hip/hip_runtime.h>
#include <stdint.h>

// Problem constants (match reference): h:[B,L,D] f32, mask:[B,M,L] i32 -> out:[B,M,D] f32
#define B_ 4
#define M_ 128
#define L_ 512
#define D_ 768
#define NEGBIG (-1e30f)

// Tiling
#define MT 16            // mention rows per block
#define DT 64            // d columns per block
#define LT 128           // l chunk per stage
#define NCH (L_ / LT)    // 4 chunks

// ---- CDNA5 async global->LDS (ASYNCcnt-tracked DMA into LDS) ----
// global_load_async_to_lds_b128: vdst = 32-bit LDS byte address, vaddr = 64-bit global address.
__device__ __forceinline__ void async_ld_b128(uint32_t lds_off, uint64_t gaddr) {
    asm volatile("global_load_async_to_lds_b128 %0, %1, off"
                 :: "v"(lds_off), "v"(gaddr)
                 : "memory");
}
__device__ __forceinline__ void wait_async0() {
    asm volatile("s_wait_asynccnt 0" ::: "memory");
}

__global__ __launch_bounds__(256)
void mention_repr_kernel(const float* __restrict__ h,
                         const int*   __restrict__ mask,
                         float*       __restrict__ out) {
    // Double-buffered LDS staging: 2*32KB h + 2*8KB mask = 80KB (<= 320KB/WGP)
    __shared__ float hT[2][LT * DT];
    __shared__ int   mkT[2][MT * LT];

    const int t  = threadIdx.x;          // 0..255
    const int d0 = blockIdx.x * DT;
    const int m0 = blockIdx.y * MT;
    const int b  = blockIdx.z;

    // Thread -> output mapping: one mention row, 4 consecutive d's (float4 lane)
    const int dloc = (t & 15) * 4;       // 0,4,...,60
    const int mloc = t >> 4;             // 0..15

    const uint64_t hbase = (uint64_t)(uintptr_t)h +
                           ((uint64_t)b * L_ * D_ + (uint64_t)d0) * 4ull;
    const uint64_t mbase = (uint64_t)(uintptr_t)mask +
                           ((uint64_t)(b * M_ + m0) * L_) * 4ull;

    // LDS byte offsets: generic LDS pointers carry the LDS offset in the low 32 bits.
    const uint32_t ldsH0 = (uint32_t)(uintptr_t)&hT[0][0];
    const uint32_t ldsH1 = (uint32_t)(uintptr_t)&hT[1][0];
    const uint32_t ldsM0 = (uint32_t)(uintptr_t)&mkT[0][0];
    const uint32_t ldsM1 = (uint32_t)(uintptr_t)&mkT[1][0];

    auto prefetch = [&](int c, int p) {
        const uint32_t ldsH = p ? ldsH1 : ldsH0;
        const uint32_t ldsM = p ? ldsM1 : ldsM0;
        const uint64_t hg = hbase + (uint64_t)c * LT * (D_ * 4ull); // chunk row start
        const uint64_t mg = mbase + (uint64_t)c * (LT * 4ull);      // chunk col start

        // h chunk: LT x DT floats = 32KB. Tile row = DT*4 = 256B contiguous in global (along d).
        // 8 issues x 256 lanes x 16B = 32KB.
#pragma unroll
        for (int j = 0; j < (LT * DT * 4) / (256 * 16); ++j) {
            const uint32_t off = (uint32_t)(j * 256 + t) * 16u;  // byte offset in tile
            const uint32_t row = off >> 8;                       // /256B
            const uint32_t col = off & 255u;
            async_ld_b128(ldsH + off, hg + (uint64_t)row * (D_ * 4ull) + col);
        }
        // mask chunk: MT x LT ints = 8KB. Tile row = LT*4 = 512B contiguous in global (along l).
        // 2 issues x 256 lanes x 16B = 8KB.
#pragma unroll
        for (int j = 0; j < (MT * LT * 4) / (256 * 16); ++j) {
            const uint32_t off = (uint32_t)(j * 256 + t) * 16u;
            const uint32_t row = off >> 9;                       // /512B
            const uint32_t col = off & 511u;
            async_ld_b128(ldsM + off, mg + (uint64_t)row * (L_ * 4ull) + col);
        }
    };

    float4 acc = make_float4(-INFINITY, -INFINITY, -INFINITY, -INFINITY);

    // Stage chunk 0, then pipeline: prefetch c+1 while reducing chunk c.
    prefetch(0, 0);
    wait_async0();
    __syncthreads();

    for (int c = 0; c < NCH; ++c) {
        const int p = c & 1;
        if (c + 1 < NCH) prefetch(c + 1, p ^ 1);

        const float* hp = &hT[p][dloc];        // lane-private 16B column slice
        const int*   mp = &mkT[p][mloc * LT];  // half-wave (16-lane) uniform broadcast row

#pragma unroll 8
        for (int l = 0; l < LT; ++l) {
            const float4 hv  = *(const float4*)(hp + l * DT);     // ds_load_b128
            const float  sel = mp[l] ? 0.0f : NEGBIG;             // ds_load_b32 + v_cndmask
            acc.x = fmaxf(acc.x, hv.x + sel);
            acc.y = fmaxf(acc.y, hv.y + sel);
            acc.z = fmaxf(acc.z, hv.z + sel);
            acc.w = fmaxf(acc.w, hv.w + sel);
        }

        wait_async0();      // this wave's chunk c+1 DMA landed in LDS
        __syncthreads();    // all waves landed + all waves done reading buffer p
    }

    float* op = out + ((uint64_t)(b * M_ + (m0 + mloc)) * D_ + (uint64_t)(d0 + dloc));
    *(float4*)op = acc;     // coalesced global_store_b128
}

extern "C" void kernel_launch(void* const* d_in, const int* in_sizes, int n_in,
                              void* d_out, int out_size, void* d_ws, size_t ws_size,
                              hipStream_t stream) {
    const float* h    = (const float*)d_in[0];
    const int*   mask = (const int*)d_in[1];
    float*       out  = (float*)d_out;

    dim3 grid(D_ / DT, M_ / MT, B_);   // (12, 8, 4) = 384 workgroups
    mention_repr_kernel<<<grid, 256, 0, stream>>>(h, mask, out);
}